// Hamilton_V8_17076789969622
// MI455X (gfx1250) — compile-verified
//
#include <hip/hip_runtime.h>
#include <math.h>

typedef __attribute__((ext_vector_type(2))) float v2f;
typedef __attribute__((ext_vector_type(8))) float v8f;

#define NROWS 2048
#define D2 256
#define D3 384

// Generic f32 WMMA GEMM: C[n,m] = epi( sum_k A[n,k] * B(k,m) )
//   BT=1  : B stored as [M,K] row-major, B(k,m) = Bs[m*ldb + k]  (contiguous v2 load)
//   BT=0  : B stored as [K,M] row-major, B(k,m) = B[k*ldb + m]
//   MODE=1: epi(x) = (1 - tanh(x + bias[m])^2) * (scale ? scale[m] : 1)
//   MODE=0: epi(x) = x
//   emul  : optional elementwise multiplier E[n,m] (same ldc), applied last
template <int BT, int MODE>
__global__ void gemm_epi(const float* __restrict__ A, int lda,
                         const float* __restrict__ B, int ldb,
                         const float* __restrict__ bias,
                         const float* __restrict__ scale,
                         const float* __restrict__ emul,
                         float* __restrict__ C, int ldc, int K) {
  const int lane = threadIdx.x;     // 0..31, one wave per block
  const int half = lane >> 4;       // 0 or 1
  const int l16  = lane & 15;
  const int row  = blockIdx.y * 16 + l16;   // A row this lane loads
  const int col  = blockIdx.x * 16 + l16;   // B col this lane loads

  v8f acc = {};
  for (int kk = 0; kk < K; kk += 4) {
    const int k0 = kk + 2 * half;
    v2f a = *(const v2f*)(A + row * lda + k0);
    v2f b;
    if (BT) {
      b = *(const v2f*)(B + col * ldb + k0);
    } else {
      b.x = B[k0 * ldb + col];
      b.y = B[(k0 + 1) * ldb + col];
    }
    acc = __builtin_amdgcn_wmma_f32_16x16x4_f32(false, a, false, b,
                                                (short)0, acc, false, false);
  }

#pragma unroll
  for (int r = 0; r < 8; ++r) {
    const int orow = blockIdx.y * 16 + r + 8 * half;
    float v = acc[r];
    if (MODE == 1) {
      float t = tanhf(v + bias[col]);
      v = 1.0f - t * t;
      if (scale) v *= scale[col];
    }
    if (emul) v *= emul[orow * ldc + col];
    C[orow * ldc + col] = v;
  }
}

// out[n,i] = sum_k P[n,k]*W1[k,i] - sum_k Q[n,k]*W2[i,k]
//   P,Q: [2048,384]; W1: [384,256] (used plain); W2: [256,384] (used transposed)
__global__ void gemm_diff(const float* __restrict__ P,
                          const float* __restrict__ W1,
                          const float* __restrict__ Q,
                          const float* __restrict__ W2,
                          float* __restrict__ Out) {
  const int lane = threadIdx.x;
  const int half = lane >> 4;
  const int l16  = lane & 15;
  const int row  = blockIdx.y * 16 + l16;
  const int col  = blockIdx.x * 16 + l16;

  v8f acc = {};
  for (int kk = 0; kk < D3; kk += 4) {
    const int k0 = kk + 2 * half;
    // + P @ W1 (W1 is [K=384, M=256] row-major, plain)
    v2f a1 = *(const v2f*)(P + row * D3 + k0);
    v2f b1;
    b1.x = W1[k0 * D2 + col];
    b1.y = W1[(k0 + 1) * D2 + col];
    acc = __builtin_amdgcn_wmma_f32_16x16x4_f32(false, a1, false, b1,
                                                (short)0, acc, false, false);
    // - Q @ W2^T  (negate A operand; W2 is [256,384] row-major -> contiguous)
    v2f a2 = *(const v2f*)(Q + row * D3 + k0);
    a2.x = -a2.x;
    a2.y = -a2.y;
    v2f b2 = *(const v2f*)(W2 + col * D3 + k0);
    acc = __builtin_amdgcn_wmma_f32_16x16x4_f32(false, a2, false, b2,
                                                (short)0, acc, false, false);
  }

#pragma unroll
  for (int r = 0; r < 8; ++r) {
    const int orow = blockIdx.y * 16 + r + 8 * half;
    Out[orow * D2 + col] = acc[r];
  }
}

extern "C" void kernel_launch(void* const* d_in, const int* in_sizes, int n_in,
                              void* d_out, int out_size, void* d_ws, size_t ws_size,
                              hipStream_t stream) {
  // setup_inputs order: t, input_, H_w1, H_b1, H_w2, H_b2, W_w1, W_b1, W_w2, W_b2
  const float* input = (const float*)d_in[1];  // [2048,256]
  const float* H_w1  = (const float*)d_in[2];  // [256,256]
  const float* H_b1  = (const float*)d_in[3];  // [256]
  const float* H_w2  = (const float*)d_in[4];  // [1,256] -> flat 256
  const float* W_w1  = (const float*)d_in[6];  // [384,256]
  const float* W_b1  = (const float*)d_in[7];  // [384]
  const float* W_w2  = (const float*)d_in[8];  // [256,384]
  float* out = (float*)d_out;                  // [2048,256]

  float* ws    = (float*)d_ws;
  float* g     = ws;                          // [2048,256]  (sH * H_w2 row)
  float* sW    = g + NROWS * D2;              // [2048,384]
  float* Hgrad = sW + NROWS * D3;             // [2048,256]
  float* P     = Hgrad + NROWS * D2;          // [2048,384]  sW .* (Hgrad@W_w2)
  float* Q     = P + NROWS * D3;              // [2048,384]  sW .* (Hgrad@W_w1^T)

  dim3 blk(32);
  dim3 g256(D2 / 16, NROWS / 16);  // (16,128)
  dim3 g384(D3 / 16, NROWS / 16);  // (24,128)

  // 1) g = (1 - tanh^2(input @ H_w1^T + H_b1)) * H_w2
  gemm_epi<1, 1><<<g256, blk, 0, stream>>>(input, D2, H_w1, D2, H_b1, H_w2,
                                           nullptr, g, D2, D2);
  // 2) sW = 1 - tanh^2(input @ W_w1^T + W_b1)
  gemm_epi<1, 1><<<g384, blk, 0, stream>>>(input, D2, W_w1, D2, W_b1, nullptr,
                                           nullptr, sW, D3, D2);
  // 3) Hgrad = g @ H_w1
  gemm_epi<0, 0><<<g256, blk, 0, stream>>>(g, D2, H_w1, D2, nullptr, nullptr,
                                           nullptr, Hgrad, D2, D2);
  // 4) P = sW .* (Hgrad @ W_w2)
  gemm_epi<0, 0><<<g384, blk, 0, stream>>>(Hgrad, D2, W_w2, D3, nullptr, nullptr,
                                           sW, P, D3, D2);
  // 5) Q = sW .* (Hgrad @ W_w1^T)
  gemm_epi<1, 0><<<g384, blk, 0, stream>>>(Hgrad, D2, W_w1, D2, nullptr, nullptr,
                                           sW, Q, D3, D2);
  // 6) out = P @ W_w1 - Q @ W_w2^T
  gemm_diff<<<g256, blk, 0, stream>>>(P, W_w1, Q, W_w2, out);
}